// GCN_Rel_73839077752936
// MI455X (gfx1250) — compile-verified
//
#include <hip/hip_runtime.h>
#include <stdint.h>

#define D_FEAT 128
#define WAVES_PER_BLOCK 8

// ---------------------------------------------------------------- zero init
// out is N*128 floats (divisible by 4) -> float4 stores; deg is scalar.
__global__ void zero_k(float4* __restrict__ a4, int na4, float* __restrict__ b, int nb) {
    int i  = blockIdx.x * blockDim.x + threadIdx.x;
    int st = gridDim.x * blockDim.x;
    const float4 z = {0.f, 0.f, 0.f, 0.f};
    for (int k = i; k < na4; k += st) a4[k] = z;
    for (int k = i; k < nb; k += st) b[k] = 0.0f;
}

// ---------------------------------------------------------------- degree count
__global__ void deg_k(const int* __restrict__ dst, float* __restrict__ deg, int E) {
    int i = blockIdx.x * blockDim.x + threadIdx.x;
    if (i < E) atomicAdd(&deg[dst[i]], 1.0f);
}

// ---------------------------------------------------------------- deg^{-1/2}
__global__ void dinv_k(const float* __restrict__ deg, float* __restrict__ dinv, int N) {
    int i = blockIdx.x * blockDim.x + threadIdx.x;
    if (i < N) {
        float d = deg[i];
        dinv[i] = (d > 0.0f) ? rsqrtf(d) : 0.0f;
    }
}

// ---------------------------------------------------------------- main scatter
// One edge per wave32: lane l owns features [4l, 4l+4). The next edge's x-row
// is DMA'd into LDS (global_load_async_to_lds_b128, saddr form, ASYNCcnt)
// while the current edge's fp32 atomics drain to L2. All per-edge state is
// forced wave-uniform via readfirstlane so edge/dinv reads ride the scalar
// cache (SMEM) and the vector memory pipe is reserved for DMA + atomics.
__global__ void __launch_bounds__(256) scatter_k(const float* __restrict__ x,
                                                 const int*   __restrict__ es,
                                                 const int*   __restrict__ ed,
                                                 const float* __restrict__ dinv,
                                                 float*       __restrict__ out,
                                                 int E) {
    __shared__ __align__(16) float lbuf[WAVES_PER_BLOCK][2][D_FEAT];

    const int lane = threadIdx.x & 31;
    const int wid  = threadIdx.x >> 5;
    // wave id is physically uniform across the wave; readfirstlane makes the
    // compiler's divergence analysis agree, scalarizing everything downstream.
    const int wave = __builtin_amdgcn_readfirstlane(blockIdx.x * WAVES_PER_BLOCK + wid);
    const int nwav = gridDim.x * WAVES_PER_BLOCK;

    float* myb = &lbuf[wid][0][0];
    // Generic pointer to LDS: low 32 bits are the workgroup-relative LDS
    // byte offset -> exactly what the async-DMA VDST VGPR wants.
    const uint32_t lds_lane = (uint32_t)(uintptr_t)myb + (uint32_t)(lane * 16);
    const uint32_t voff     = (uint32_t)(lane * 16);   // per-lane 16B slice

    int e = wave;
    if (e >= E) return;  // wave-uniform exit: EXEC stays all-ones inside

    int s0 = es[e], d0 = ed[e];
    {   // prime the pipeline: stage row x[s0] -> buffer 0
        uint64_t gb = (uint64_t)(uintptr_t)(x + (size_t)s0 * D_FEAT);
        asm volatile("global_load_async_to_lds_b128 %0, %1, %2"
                     :: "v"(lds_lane), "v"(voff), "s"(gb) : "memory");
    }

    int p = 0;
    for (;;) {
        const int  en = e + nwav;
        const bool hn = (en < E);
        int s1 = 0, d1 = 0;
        if (hn) {
            s1 = es[en];                                 // s_load_b32 (SMEM)
            d1 = ed[en];
            __builtin_prefetch(&es[en + nwav], 0, 0);    // prefetch edge stream
            // ensure previous ds_load of the target buffer fully retired
            asm volatile("s_wait_dscnt 0x0" ::: "memory");
            uint64_t gb = (uint64_t)(uintptr_t)(x + (size_t)s1 * D_FEAT);
            uint32_t la = lds_lane + (uint32_t)((p ^ 1) * (D_FEAT * 4));
            asm volatile("global_load_async_to_lds_b128 %0, %1, %2"
                         :: "v"(la), "v"(voff), "s"(gb) : "memory");
            asm volatile("s_wait_asynccnt 0x1" ::: "memory");  // current buf landed
        } else {
            asm volatile("s_wait_asynccnt 0x0" ::: "memory");  // drain tail
        }

        const float norm = dinv[s0] * dinv[d0];          // scalar: s_load + s_mul_f32
        const float4 v   = ((const float4*)(myb + p * D_FEAT))[lane];  // ds_load_b128

        float* ob = out + (size_t)d0 * D_FEAT + (size_t)(lane * 4);  // saddr + voffset
        atomicAdd(ob + 0, v.x * norm);   // global_atomic_add_f32, non-returning
        atomicAdd(ob + 1, v.y * norm);
        atomicAdd(ob + 2, v.z * norm);
        atomicAdd(ob + 3, v.w * norm);

        if (!hn) break;
        e = en; s0 = s1; d0 = d1; p ^= 1;
    }
}

// ---------------------------------------------------------------- ReLU (float4)
__global__ void relu_k(float4* __restrict__ out4, int n4) {
    int i  = blockIdx.x * blockDim.x + threadIdx.x;
    int st = gridDim.x * blockDim.x;
    for (int k = i; k < n4; k += st) {
        float4 v = out4[k];
        v.x = fmaxf(v.x, 0.f);
        v.y = fmaxf(v.y, 0.f);
        v.z = fmaxf(v.z, 0.f);
        v.w = fmaxf(v.w, 0.f);
        out4[k] = v;
    }
}

// ---------------------------------------------------------------- launch
extern "C" void kernel_launch(void* const* d_in, const int* in_sizes, int n_in,
                              void* d_out, int out_size, void* d_ws, size_t ws_size,
                              hipStream_t stream) {
    const float* x  = (const float*)d_in[0];
    const int*   ei = (const int*)d_in[1];   // [2, E] row-major: row0=src(j), row1=dst(i)
    // d_in[2] (line_graph_val) unused by the reference module.

    const int NF = in_sizes[0];          // N * 128
    const int N  = NF / D_FEAT;
    const int E  = in_sizes[1] / 2;

    const int* src = ei;
    const int* dst = ei + E;

    float* out  = (float*)d_out;
    float* deg  = (float*)d_ws;          // N floats
    float* dinv = deg + N;               // N floats

    // 1) zero accumulator (vectorized) + degree
    zero_k<<<2048, 256, 0, stream>>>((float4*)out, NF / 4, deg, N);
    // 2) degree = scatter-add of ones over dst
    deg_k<<<(E + 255) / 256, 256, 0, stream>>>(dst, deg, E);
    // 3) deg^{-1/2} with zero-guard
    dinv_k<<<(N + 255) / 256, 256, 0, stream>>>(deg, dinv, N);
    // 4) normalized scatter aggregation: 32768 waves, ~18 edges each,
    //    async-LDS double buffered, scalar edge stream
    scatter_k<<<4096, 256, 0, stream>>>(x, src, dst, dinv, out, E);
    // 5) ReLU in place (vectorized)
    relu_k<<<2048, 256, 0, stream>>>((float4*)out, NF / 4);
}